// ResamplerLayer_11596411699350
// MI455X (gfx1250) — compile-verified
//
#include <hip/hip_runtime.h>
#include <hip/hip_bf16.h>

// Trilinear resampler for MI455X (gfx1250, wave32).
// inputs:        [B=2, D=128, H=128, W=128, C=4]  float32
// sample_coords: [B, 96, 96, 96, 3]               float32 (z,y,x order = dims 0,1,2)
// out:           [B, 96, 96, 96, 4]               float32
//
// Gather-bound op. Input volume (67 MB) fits in the 192 MB L2 -> corner
// gathers use default (RT) temporal policy so the volume stays L2-resident;
// the write-once output stream uses non-temporal stores to avoid evicting it.

typedef float v4f __attribute__((ext_vector_type(4)));

#define R_B   2
#define R_ID  128
#define R_IH  128
#define R_IW  128
#define R_OD  96
#define R_OH  96
#define R_OW  96

static constexpr int PTS_PER_BATCH = R_OD * R_OH * R_OW;        // 884736
static constexpr int NPTS          = R_B * PTS_PER_BATCH;       // 1769472

__device__ __forceinline__ v4f blend(v4f even, v4f odd, float w1, float w0) {
    // reference: f = samples_even * w_1 + samples_odd * w_0
    return even * w1 + odd * w0;
}

__global__ __launch_bounds__(256)
void ResamplerLayer_11596411699350_kernel(const float* __restrict__ inputs,
                                          const float* __restrict__ coords,
                                          float* __restrict__ out) {
    const int i = blockIdx.x * blockDim.x + threadIdx.x;
    if (i >= NPTS) return;

    // ---- coords (3 consecutive floats per point -> coalesced b96 load) ----
    const float* cp = coords + (size_t)i * 3;
    const float cz = cp[0];
    const float cy = cp[1];
    const float cx = cp[2];

    // ---- faithful clamp semantics (clip AFTER int conversion, weights from
    //      the clamped corner coordinates) ----
    const float fz = floorf(cz);
    const float fy = floorf(cy);
    const float fx = floorf(cx);

    int z0 = (int)fz;          int z1 = (int)(fz + 1.0f);
    int y0 = (int)fy;          int y1 = (int)(fy + 1.0f);
    int x0 = (int)fx;          int x1 = (int)(fx + 1.0f);

    z0 = min(max(z0, 0), R_ID - 1);  z1 = min(max(z1, 0), R_ID - 1);
    y0 = min(max(y0, 0), R_IH - 1);  y1 = min(max(y1, 0), R_IH - 1);
    x0 = min(max(x0, 0), R_IW - 1);  x1 = min(max(x1, 0), R_IW - 1);

    const float w0z = cz - (float)z0;   const float w1z = (float)z1 - cz;
    const float w0y = cy - (float)y0;   const float w1y = (float)y1 - cy;
    const float w0x = cx - (float)x0;   const float w1x = (float)x1 - cx;

    // ---- gather 8 corners, each one float4 (global_load_b128) ----
    const int b = i / PTS_PER_BATCH;            // B==2, strength-reduced by compiler
    const v4f* __restrict__ in4 = (const v4f*)inputs;

    const unsigned base = (unsigned)b * (unsigned)(R_ID * R_IH * R_IW);
    const unsigned zo0  = base + (unsigned)z0 * (unsigned)(R_IH * R_IW);
    const unsigned zo1  = base + (unsigned)z1 * (unsigned)(R_IH * R_IW);
    const unsigned yo0  = (unsigned)y0 * (unsigned)R_IW;
    const unsigned yo1  = (unsigned)y1 * (unsigned)R_IW;

    const v4f s000 = in4[zo0 + yo0 + (unsigned)x0];
    const v4f s001 = in4[zo0 + yo0 + (unsigned)x1];
    const v4f s010 = in4[zo0 + yo1 + (unsigned)x0];
    const v4f s011 = in4[zo0 + yo1 + (unsigned)x1];
    const v4f s100 = in4[zo1 + yo0 + (unsigned)x0];
    const v4f s101 = in4[zo1 + yo0 + (unsigned)x1];
    const v4f s110 = in4[zo1 + yo1 + (unsigned)x0];
    const v4f s111 = in4[zo1 + yo1 + (unsigned)x1];

    // ---- pyramid combination, matching reference recursion exactly:
    //      innermost over x (last weight), then y, then z. ----
    const v4f t00 = blend(s000, s001, w1x, w0x);
    const v4f t01 = blend(s010, s011, w1x, w0x);
    const v4f t10 = blend(s100, s101, w1x, w0x);
    const v4f t11 = blend(s110, s111, w1x, w0x);

    const v4f u0 = blend(t00, t01, w1y, w0y);
    const v4f u1 = blend(t10, t11, w1y, w0y);

    const v4f r = blend(u0, u1, w1z, w0z);

    // ---- streamed write-once output: non-temporal b128 store so the 28 MB
    //      output stream does not evict the L2-resident input volume ----
    __builtin_nontemporal_store(r, (v4f*)out + i);
}

extern "C" void kernel_launch(void* const* d_in, const int* in_sizes, int n_in,
                              void* d_out, int out_size, void* d_ws, size_t ws_size,
                              hipStream_t stream) {
    (void)in_sizes; (void)n_in; (void)d_ws; (void)ws_size; (void)out_size;
    const float* inputs = (const float*)d_in[0];
    const float* coords = (const float*)d_in[1];
    float* out = (float*)d_out;

    const int threads = 256;
    const int blocks  = (NPTS + threads - 1) / threads;   // 6912
    ResamplerLayer_11596411699350_kernel<<<blocks, threads, 0, stream>>>(inputs, coords, out);
}